// GatedGraphConv_86277303042060
// MI455X (gfx1250) — compile-verified
//
#include <hip/hip_runtime.h>
#include <math.h>

typedef float v2f __attribute__((ext_vector_type(2)));
typedef float v8f __attribute__((ext_vector_type(8)));

#define DD 128
#define THREE_D 384

__global__ void zero_f32(float* __restrict__ p, int n) {
  int i = blockIdx.x * blockDim.x + threadIdx.x;
  if (i < n) p[i] = 0.0f;
}

// C[16-row panel, 128-col panel] (+)= A[rows x 128] @ W[128 x cols]
// A: row-major, lda = 128.  W: row-major, leading dim ldw, columns [wcol ...).
// C: row-major, leading dim ldc, columns [ccol ...).
// grid.x = N/16 row tiles, grid.y = cols/128, block = 256 threads (8 waves),
// wave w handles a 16x16 tile. Operands staged in LDS via float4 so the
// unrolled inner loop is v_wmma_f32_16x16x4_f32 + ds_load immediate-offset.
// All call sites guarantee 16B alignment (lda=128, ldw in {128,384},
// wcol in {0,256}, 128-col panels).
__global__ void gemm128_wmma(const float* __restrict__ A,
                             const float* __restrict__ W, int ldw, int wcol,
                             float* __restrict__ C, int ldc, int ccol,
                             int accumulate) {
  __shared__ float As[16][132];  // stride 132 (x4B=528, 16B-aligned rows)
  __shared__ float Ws[32][136];  // stride 136 (x4B=544, 16B-aligned rows)

  const int tid = threadIdx.x;
  const int rowBase = blockIdx.x * 16;

  // Cooperative A panel load: 512 float4s, 2 per thread (coalesced b128).
#pragma unroll
  for (int i = 0; i < 2; ++i) {
    int flat = i * 256 + tid;  // 0..511
    int r = flat >> 5;         // 0..15
    int c4 = (flat & 31) * 4;  // 0,4,...,124
    float4 v = *(const float4*)(A + (long)(rowBase + r) * DD + c4);
    *(float4*)&As[r][c4] = v;
  }

  const int wave = tid >> 5;
  const int lane = tid & 31;
  const int hl = lane >> 4;           // half-wave select
  const int mn = lane & 15;           // M (for A) / N (for B,C)
  const int koff = hl * 2;            // lanes 16-31 hold K+2,K+3
  const int wbase = wave * 16 + mn;   // column within the 128-col panel
  const int colC = ccol + blockIdx.y * 128 + wave * 16;

  // W staging slot: row tid>>5 (0..7), 4 cols at (tid&31)*4. Each of the 4
  // unrolled steps advances 8 rows -> short running-pointer chain.
  const int wr = tid >> 5;
  const int wc4 = (tid & 31) * 4;
  const float* wload =
      W + (long)wr * ldw + wcol + blockIdx.y * 128 + wc4;

  v8f acc;
#pragma unroll
  for (int i = 0; i < 8; ++i) {
    acc[i] = accumulate
                 ? C[(long)(rowBase + i + 8 * hl) * ldc + colC + mn]
                 : 0.0f;
  }

  for (int kc = 0; kc < 128; kc += 32) {
    __syncthreads();  // also covers the initial As writes on first pass
    // Stage 32x128 W chunk into LDS: 4 x b128 per thread, coalesced.
    const float* p = wload + (long)kc * ldw;
#pragma unroll
    for (int i = 0; i < 4; ++i) {
      float4 v = *(const float4*)p;
      *(float4*)&Ws[wr + 8 * i][wc4] = v;
      p += 8 * (long)ldw;
    }
    __syncthreads();

#pragma unroll
    for (int k = 0; k < 32; k += 4) {
      // A fragment: 16x4 f32. lanes 0-15: A[mn][k..k+1]; 16-31: A[mn][k+2..3]
      v2f a;
      a.x = As[mn][kc + k + koff];
      a.y = As[mn][kc + k + koff + 1];
      // B fragment: 4x16 f32. VGPR j, lanes 0-15: K=k+j; lanes 16-31: K=k+j+2
      v2f b;
      b.x = Ws[k + koff][wbase];
      b.y = Ws[k + koff + 1][wbase];
      acc = __builtin_amdgcn_wmma_f32_16x16x4_f32(
          /*neg_a=*/false, a, /*neg_b=*/false, b,
          /*c_mod=*/(short)0, acc, /*reuse_a=*/false, /*reuse_b=*/false);
    }
  }

#pragma unroll
  for (int i = 0; i < 8; ++i)
    C[(long)(rowBase + i + 8 * hl) * ldc + colC + mn] = acc[i];
}

// agg[receivers[e]][d] += m[senders[e]][d]  (one float atomic per edge-channel)
__global__ void scatter_add(const float* __restrict__ m,
                            const int* __restrict__ senders,
                            const int* __restrict__ receivers,
                            float* __restrict__ agg, int total) {
  int idx = blockIdx.x * blockDim.x + threadIdx.x;
  if (idx >= total) return;
  int e = idx >> 7;
  int d = idx & 127;
  int s = senders[e];
  int r = receivers[e];
  atomicAdd(agg + (long)r * DD + d, m[(long)s * DD + d]);
}

// zr = sigmoid(gx[:, :256] + b[:256]); store z back; rx = r * x
__global__ void gru_part1(float* __restrict__ gx, const float* __restrict__ b,
                          const float* __restrict__ x, float* __restrict__ rx,
                          int n) {
  int idx = blockIdx.x * blockDim.x + threadIdx.x;
  if (idx >= n) return;
  int row = idx >> 7;
  int d = idx & 127;
  long g = (long)row * THREE_D;
  float z = 1.0f / (1.0f + expf(-(gx[g + d] + b[d])));
  float r = 1.0f / (1.0f + expf(-(gx[g + 128 + d] + b[128 + d])));
  gx[g + d] = z;
  rx[idx] = r * x[idx];
}

// a = tanh(gx[:,256:] + b[256:]); out = (1-z)*x + z*a
__global__ void gru_part2(const float* __restrict__ gx,
                          const float* __restrict__ b,
                          const float* __restrict__ x, float* __restrict__ out,
                          int n) {
  int idx = blockIdx.x * blockDim.x + threadIdx.x;
  if (idx >= n) return;
  int row = idx >> 7;
  int d = idx & 127;
  long g = (long)row * THREE_D;
  float z = gx[g + d];
  float a = tanhf(gx[g + 256 + d] + b[256 + d]);
  float xv = x[idx];
  out[idx] = (1.0f - z) * xv + z * a;
}

extern "C" void kernel_launch(void* const* d_in, const int* in_sizes, int n_in,
                              void* d_out, int out_size, void* d_ws,
                              size_t ws_size, hipStream_t stream) {
  const float* nodes = (const float*)d_in[0];
  const int* senders = (const int*)d_in[1];
  const int* receivers = (const int*)d_in[2];
  const float* weights = (const float*)d_in[3];
  const float* w_i = (const float*)d_in[4];
  const float* w_h = (const float*)d_in[5];
  const float* bias = (const float*)d_in[6];
  float* out = (float*)d_out;

  const int N = in_sizes[0] / DD;            // 100000
  const int E = in_sizes[1];                 // 600000
  const int L = in_sizes[3] / (DD * DD);     // 3

  // workspace layout: B0 (N*128) | agg (N*128) | gx (N*384)  -> 640*N floats
  float* B0 = (float*)d_ws;
  float* agg = B0 + (long)N * DD;
  float* gx = agg + (long)N * DD;

  const dim3 blk(256);
  const int rowTiles = N / 16;               // N divisible by 16
  const int nd = N * DD;
  const int gridEls = (nd + 255) / 256;
  const int gridEdge = (E * DD + 255) / 256;

  const float* x = nodes;
  for (int l = 0; l < L; ++l) {
    float* S = (l == 1) ? out : B0;          // scratch: m, then rx
    float* xdst = (l == 0) ? B0 : out;       // next state

    // 1. m = x @ weights[l]
    gemm128_wmma<<<dim3(rowTiles, 1), blk, 0, stream>>>(
        x, weights + (long)l * DD * DD, DD, 0, S, DD, 0, 0);
    // 2. agg = 0
    zero_f32<<<gridEls, blk, 0, stream>>>(agg, nd);
    // 3. agg[r] += m[s]
    scatter_add<<<gridEdge, blk, 0, stream>>>(S, senders, receivers, agg,
                                              E * DD);
    // 4. gx = agg @ w_i                   (N x 384)
    gemm128_wmma<<<dim3(rowTiles, 3), blk, 0, stream>>>(agg, w_i, THREE_D, 0,
                                                        gx, THREE_D, 0, 0);
    // 5. gx[:, :256] += x @ w_h[:, :256]
    gemm128_wmma<<<dim3(rowTiles, 2), blk, 0, stream>>>(x, w_h, THREE_D, 0, gx,
                                                        THREE_D, 0, 1);
    // 6. z,r = sigmoid(...); rx = r * x
    gru_part1<<<gridEls, blk, 0, stream>>>(gx, bias, x, S, nd);
    // 7. gx[:, 256:] += rx @ w_h[:, 256:]
    gemm128_wmma<<<dim3(rowTiles, 1), blk, 0, stream>>>(S, w_h, THREE_D, 256,
                                                        gx, THREE_D, 256, 1);
    // 8. x_new = (1-z)*x + z*tanh(a_x + a_h + b)
    gru_part2<<<gridEls, blk, 0, stream>>>(gx, bias, x, xdst, nd);
    x = xdst;
  }
}